// CRD_62062277427822
// MI455X (gfx1250) — compile-verified
//
#include <hip/hip_runtime.h>

typedef __attribute__((ext_vector_type(2))) float v2f;
typedef __attribute__((ext_vector_type(8))) float v8f;

#define N_NODES 100000
#define N_EDGES 1600000
#define D_IN 128
#define D_OUT 64

// ---------------- degree / normalization ----------------

__global__ void k_init_deg(float* __restrict__ deg) {
    int i = blockIdx.x * blockDim.x + threadIdx.x;
    if (i < N_NODES) deg[i] = 1.0f;  // self-loop contributes 1 to every node
}

__global__ void k_edge_degree(const int* __restrict__ dst, float* __restrict__ deg) {
    int e = blockIdx.x * blockDim.x + threadIdx.x;
    if (e < N_EDGES) atomicAdd(&deg[dst[e]], 1.0f);
}

__global__ void k_dinv(float* __restrict__ deg) {
    int i = blockIdx.x * blockDim.x + threadIdx.x;
    if (i < N_NODES) {
        float d = deg[i];
        deg[i] = (d > 0.0f) ? rsqrtf(d) : 0.0f;   // in-place: deg -> dinv
    }
}

// ---------------- GEMM h = X*W via V_WMMA_F32_16X16X4_F32 ----------------
// One wave computes a 16x64 output tile (4 n-tiles of 16x16, K swept 4 at a time).
// Epilogue: out = h * dinv[row]^2 + bias  (self-loop term folded in), h stored raw.

__global__ __launch_bounds__(256) void k_gemm_wmma(
    const float* __restrict__ x, const float* __restrict__ w,
    const float* __restrict__ bias, const float* __restrict__ dinv,
    float* __restrict__ h, float* __restrict__ out) {

    __shared__ float wlds[D_IN * D_OUT];  // 32 KB: W staged once per block
    const int tid = threadIdx.x;
    for (int i = tid; i < D_IN * D_OUT; i += 256) wlds[i] = w[i];
    __syncthreads();

    const int wave = tid >> 5;
    const int lane = tid & 31;
    const int rowbase = (blockIdx.x * 8 + wave) * 16;
    if (rowbase >= N_NODES) return;  // wave-uniform exit (EXEC stays all-1 for WMMA)

    const int l15 = lane & 15;
    const int kk  = (lane < 16) ? 0 : 2;  // f32 A/B frag: lanes 16-31 hold K+2
    const float* xrow = x + (long)(rowbase + l15) * D_IN + kk;

    v8f acc[4];
    acc[0] = (v8f){}; acc[1] = (v8f){}; acc[2] = (v8f){}; acc[3] = (v8f){};

#pragma unroll
    for (int k0 = 0; k0 < D_IN; k0 += 4) {
        // A (16x4): lane = M row, holds K = kk, kk+1
        v2f a;
        a.x = xrow[k0];
        a.y = xrow[k0 + 1];
        // B (4x16): lane = N col, VGPR0/1 = K kk / kk+1
        const float* wp = &wlds[(k0 + kk) * D_OUT + l15];
#pragma unroll
        for (int nt = 0; nt < 4; ++nt) {
            v2f b;
            b.x = wp[nt * 16];
            b.y = wp[nt * 16 + D_OUT];
            acc[nt] = __builtin_amdgcn_wmma_f32_16x16x4_f32(
                false, a, false, b, (short)0, acc[nt], false, false);
        }
    }

    // D layout (16x16 f32): VGPR r -> M = r (lanes 0-15) / r+8 (lanes 16-31), N = lane&15
    const int rofs = (lane < 16) ? 0 : 8;
#pragma unroll
    for (int r = 0; r < 8; ++r) {
        const int row = rowbase + rofs + r;
        const float di = dinv[row];
        const float self_n = di * di;
#pragma unroll
        for (int nt = 0; nt < 4; ++nt) {
            const int col = nt * 16 + l15;
            const float v = acc[nt][r];
            const long o = (long)row * D_OUT + col;
            h[o]   = v;                          // raw h for edge gather
            out[o] = v * self_n + bias[col];     // self-loop + bias init
        }
    }
}

// ---------------- edge scatter: out[dst] += h[src] * dinv[src]*dinv[dst] ---------

__global__ void k_scatter(const int* __restrict__ src, const int* __restrict__ dst,
                          const float* __restrict__ dinv, const float* __restrict__ h,
                          float* __restrict__ out) {
    // 16 threads per edge, each handles 4 consecutive columns (float4 gather).
    long gid = (long)blockIdx.x * blockDim.x + threadIdx.x;
    long e = gid >> 4;
    if (e >= N_EDGES) return;
    const int q = (int)(gid & 15);

    const int s = src[e];
    const int d = dst[e];
    const float scale = dinv[s] * dinv[d];

    const float4 hv = *(const float4*)(h + (long)s * D_OUT + q * 4);
    float* op = out + (long)d * D_OUT + q * 4;
    atomicAdd(op + 0, hv.x * scale);
    atomicAdd(op + 1, hv.y * scale);
    atomicAdd(op + 2, hv.z * scale);
    atomicAdd(op + 3, hv.w * scale);
}

__global__ void k_relu(float* __restrict__ out) {
    int i = blockIdx.x * blockDim.x + threadIdx.x;
    if (i < N_NODES * D_OUT) out[i] = fmaxf(out[i], 0.0f);
}

// ---------------- launch ----------------

extern "C" void kernel_launch(void* const* d_in, const int* in_sizes, int n_in,
                              void* d_out, int out_size, void* d_ws, size_t ws_size,
                              hipStream_t stream) {
    const float* x    = (const float*)d_in[0];             // [N, 128]
    const int*   ei   = (const int*)d_in[1];               // [2, E] row-major
    const float* w    = (const float*)d_in[2];             // [128, 64]
    const float* bias = (const float*)d_in[3];             // [64]
    float* out = (float*)d_out;                            // [N, 64]

    float* h    = (float*)d_ws;                            // N*64 floats
    float* dinv = h + (size_t)N_NODES * D_OUT;             // N floats (deg then dinv)

    const int* srcp = ei;
    const int* dstp = ei + N_EDGES;

    k_init_deg   <<<(N_NODES + 255) / 256, 256, 0, stream>>>(dinv);
    k_edge_degree<<<(N_EDGES + 255) / 256, 256, 0, stream>>>(dstp, dinv);
    k_dinv       <<<(N_NODES + 255) / 256, 256, 0, stream>>>(dinv);

    const int m_tiles = (N_NODES + 15) / 16;               // 6250
    k_gemm_wmma  <<<(m_tiles + 7) / 8, 256, 0, stream>>>(x, w, bias, dinv, h, out);

    const long sth = (long)N_EDGES * 16;
    k_scatter    <<<(int)((sth + 255) / 256), 256, 0, stream>>>(srcp, dstp, dinv, h, out);

    k_relu       <<<(N_NODES * D_OUT + 255) / 256, 256, 0, stream>>>(out);
}